// PAPAGATChannel_73864847557067
// MI455X (gfx1250) — compile-verified
//
#include <hip/hip_runtime.h>
#include <math.h>

// ---------------------------------------------------------------------------
// 2-layer GATConv forward for MI455X (gfx1250, wave32).
// GEMMs use V_WMMA_F32_16X16X4_F32 (fp32 matrix path) with the A-tile staged
// into LDS via GLOBAL_LOAD_ASYNC_TO_LDS_B128 (ASYNCcnt-tracked CDNA5 async
// copy). Edge softmax uses ordered-uint atomicMax for segment-max and f32
// global atomics for the unnormalized weighted scatter-sum (L2-resident:
// h1 = 51MB << 192MB L2).
// ---------------------------------------------------------------------------

typedef __attribute__((ext_vector_type(2))) float v2f;
typedef __attribute__((ext_vector_type(8))) float v8f;

#if __has_builtin(__builtin_amdgcn_global_load_async_to_lds_b128) && \
    __has_builtin(__builtin_amdgcn_s_wait_asynccnt)
#define GAT_ASYNC_LDS 1
// Builtin signature (per hipcc diagnostic): pointee is a GCC-vector of 4 ints.
typedef int v4i_ __attribute__((vector_size(16)));
typedef __attribute__((address_space(1))) v4i_ gv4i;   // global
typedef __attribute__((address_space(3))) v4i_ lv4i;   // LDS
#endif

// Monotonic float <-> uint mapping so atomicMax(uint) == float max.
__device__ __forceinline__ unsigned fenc(float f) {
    unsigned u = __float_as_uint(f);
    return (u & 0x80000000u) ? ~u : (u | 0x80000000u);
}
__device__ __forceinline__ float fdec(unsigned u) {
    unsigned b = (u & 0x80000000u) ? (u & 0x7fffffffu) : ~u;
    return __uint_as_float(b);
}

// ---------------------------------------------------------------------------
// WMMA f32 GEMM: C[M,Ncols] = A[M,K] * B[K,Ncols].
// Block = 256 threads = 8 waves; block tile = (rowTiles*16) x (colTiles*32),
// rowTiles*colTiles == 8. Each wave owns a 16x32 tile: one A fragment feeds
// two 16x16 WMMA accumulators (halves A traffic, doubles WMMA density).
// Per-lane fragment layouts per CDNA5 ISA 7.12.2:
//   A 16x4 f32 : lanes 0-15 hold K={0,1}, lanes 16-31 hold K={2,3} (rows = lane%16)
//   B 4x16 f32 : mirrored (cols = lane%16)
//   C 16x16 f32: VGPR i -> row i (lanes 0-15) / row i+8 (lanes 16-31), col = lane%16
// ---------------------------------------------------------------------------
__global__ void gat_wmma_gemm_f32(const float* __restrict__ A,
                                  const float* __restrict__ B,
                                  float* __restrict__ C,
                                  int M, int K, int Ncols,
                                  int rowTiles, int colTiles) {
    extern __shared__ float ldsA[];
    const int tid  = threadIdx.x;
    const int lane = tid & 31;
    const int wave = tid >> 5;

    const int rowsInBlk = rowTiles * 16;
    const int rowBase   = blockIdx.x * rowsInBlk;

#ifdef GAT_ASYNC_LDS
    // CDNA5 async copy: 16B per lane, global -> LDS, no VGPR round trip.
    // Tracked by ASYNCcnt; each wave drains its own count before the barrier.
    for (int i = tid * 4; i < rowsInBlk * K; i += (int)blockDim.x * 4) {
        int r = i / K, c = i - r * K;          // K % 4 == 0 -> 16B stays in-row
        int gr = rowBase + r;
        if (gr >= M) gr = M - 1;
        const float* gp = &A[(long long)gr * K + c];
        float* lp = &ldsA[i];
        __builtin_amdgcn_global_load_async_to_lds_b128(
            (gv4i*)(unsigned long long)gp,
            (lv4i*)(unsigned)(unsigned long long)lp,
            0, 0);
    }
    __builtin_amdgcn_s_wait_asynccnt(0);
    __syncthreads();
#else
    for (int i = tid; i < rowsInBlk * K; i += blockDim.x) {
        int r = i / K, c = i - r * K;
        int gr = rowBase + r;
        if (gr >= M) gr = M - 1;
        ldsA[i] = A[(long long)gr * K + c];
    }
    __syncthreads();
#endif

    const int colTile = wave % colTiles;
    const int rowTile = wave / colTiles;
    const int colBase = blockIdx.y * (colTiles * 32) + colTile * 32;

    const int mn   = lane & 15;          // row of A frag / col of B frag
    const int koff = (lane >> 4) * 2;    // K sub-offset per lane half
    const float* lA = &ldsA[rowTile * 16 * K];

    v8f acc0 = {0.f, 0.f, 0.f, 0.f, 0.f, 0.f, 0.f, 0.f};
    v8f acc1 = {0.f, 0.f, 0.f, 0.f, 0.f, 0.f, 0.f, 0.f};
    for (int k = 0; k < K; k += 4) {
        v2f a, b0, b1;
        a.x = lA[mn * K + k + koff];
        a.y = lA[mn * K + k + koff + 1];
        const float* Brow0 = &B[(long long)(k + koff)     * Ncols + colBase + mn];
        const float* Brow1 = &B[(long long)(k + koff + 1) * Ncols + colBase + mn];
        b0.x = Brow0[0];   b0.y = Brow1[0];
        b1.x = Brow0[16];  b1.y = Brow1[16];
        // (neg_a, A, neg_b, B, c_mod, C, reuse_a, reuse_b)
        acc0 = __builtin_amdgcn_wmma_f32_16x16x4_f32(
                   false, a, false, b0, (short)0, acc0, false, false);
        acc1 = __builtin_amdgcn_wmma_f32_16x16x4_f32(
                   false, a, false, b1, (short)0, acc1, false, false);
    }

    const int rowHalf = (lane >> 4) * 8;
#pragma unroll
    for (int i = 0; i < 8; ++i) {
        int r = rowBase + rowTile * 16 + rowHalf + i;
        if (r < M) {
            C[(long long)r * Ncols + colBase + mn]      = acc0[i];
            C[(long long)r * Ncols + colBase + 16 + mn] = acc1[i];
        }
    }
}

// ---------------------------------------------------------------------------
// Per-node attention logits: a_s[n,h] = <h[n,h,:], att_src[h,:]>, same for dst.
// One wave per node; lanes stride channels; butterfly shuffle reduction.
// ---------------------------------------------------------------------------
__global__ void gat_att_logits(const float* __restrict__ hfeat,
                               const float* __restrict__ att_src,
                               const float* __restrict__ att_dst,
                               float* __restrict__ a_s, float* __restrict__ a_d,
                               int N, int H, int C) {
    int wid  = (int)(((long long)blockIdx.x * blockDim.x + threadIdx.x) >> 5);
    int lane = threadIdx.x & 31;
    if (wid >= N) return;
    const float* row = hfeat + (long long)wid * H * C;
    for (int hh = 0; hh < H; ++hh) {
        float ss = 0.f, sd = 0.f;
        for (int c = lane; c < C; c += 32) {
            float v = row[hh * C + c];
            ss += v * att_src[hh * C + c];
            sd += v * att_dst[hh * C + c];
        }
        for (int off = 16; off; off >>= 1) {
            ss += __shfl_xor(ss, off, 32);
            sd += __shfl_xor(sd, off, 32);
        }
        if (lane == 0) { a_s[wid * H + hh] = ss; a_d[wid * H + hh] = sd; }
    }
}

// Init: m <- enc(-inf), denom <- 0 (first mN threads), accum <- 0 (accN threads)
__global__ void gat_init(unsigned* __restrict__ m, float* __restrict__ denom,
                         float* __restrict__ accum, int mN, int accN) {
    long long i = (long long)blockIdx.x * blockDim.x + threadIdx.x;
    if (i < mN) { m[i] = fenc(-__builtin_inff()); denom[i] = 0.f; }
    if (i < accN) accum[i] = 0.f;
}

// Pass A: segment-max of leaky_relu(a_s[src]+a_d[dst]) over dst (self loops appended).
__global__ void gat_edge_max(const int* __restrict__ ei, int E, int N, int H,
                             const float* __restrict__ a_s,
                             const float* __restrict__ a_d,
                             unsigned* __restrict__ m) {
    long long idx = (long long)blockIdx.x * blockDim.x + threadIdx.x;
    long long total = (long long)(E + N) * H;
    if (idx >= total) return;
    int e  = (int)(idx / H);
    int hh = (int)(idx - (long long)e * H);
    int src, dst;
    if (e < E) { src = ei[e]; dst = ei[E + e]; } else { src = dst = e - E; }
    float v = a_s[src * H + hh] + a_d[dst * H + hh];
    v = (v >= 0.f) ? v : 0.2f * v;              // leaky_relu(., 0.2)
    atomicMax(&m[dst * H + hh], fenc(v));
}

// Pass B: p = exp(e - m[dst]); denom[dst] += p; accum[dst,:] += p * h[src,:].
// One wave per edge; lanes cover the H*C feature row (coalesced gather+atomics).
__global__ void gat_edge_agg(const int* __restrict__ ei, int E, int N, int H, int C,
                             const float* __restrict__ a_s,
                             const float* __restrict__ a_d,
                             const unsigned* __restrict__ m,
                             const float* __restrict__ hfeat,
                             float* __restrict__ denom,
                             float* __restrict__ accum) {
    int wid  = (int)(((long long)blockIdx.x * blockDim.x + threadIdx.x) >> 5);
    int lane = threadIdx.x & 31;
    if (wid >= E + N) return;
    int src, dst;
    if (wid < E) { src = ei[wid]; dst = ei[E + wid]; } else { src = dst = wid - E; }

    float p[4];
    for (int hh = 0; hh < H; ++hh) {
        float v = a_s[src * H + hh] + a_d[dst * H + hh];
        v = (v >= 0.f) ? v : 0.2f * v;
        p[hh] = __expf(v - fdec(m[dst * H + hh]));
    }
    if (lane < H) atomicAdd(&denom[dst * H + lane], p[lane]);

    const int D = H * C;
    const float* hs = hfeat + (long long)src * D;
    __builtin_prefetch(hs, 0, 0);               // global_prefetch_b8 (CDNA5)
    float* ac = accum + (long long)dst * D;
    for (int v = lane; v < D; v += 32)
        atomicAdd(&ac[v], p[v / C] * hs[v]);
}

// accum[n,v] = accum[n,v]/denom[n, v/C] + bias[v]; optional ReLU. In place.
__global__ void gat_finalize(float* __restrict__ accum,
                             const float* __restrict__ denom,
                             const float* __restrict__ bias,
                             int N, int H, int C, int do_relu) {
    long long i = (long long)blockIdx.x * blockDim.x + threadIdx.x;
    int D = H * C;
    long long total = (long long)N * D;
    if (i >= total) return;
    int n = (int)(i / D);
    int v = (int)(i - (long long)n * D);
    float val = accum[i] / denom[n * H + v / C] + bias[v];
    if (do_relu) val = fmaxf(val, 0.f);
    accum[i] = val;
}

// Row-wise L2 normalize of [N,64] -> d_out. One wave per node (2 elems/lane).
__global__ void gat_l2norm(const float* __restrict__ in, float* __restrict__ out, int N) {
    int wid  = (int)(((long long)blockIdx.x * blockDim.x + threadIdx.x) >> 5);
    int lane = threadIdx.x & 31;
    if (wid >= N) return;
    float v0 = in[wid * 64 + lane];
    float v1 = in[wid * 64 + 32 + lane];
    float s = v0 * v0 + v1 * v1;
    for (int off = 16; off; off >>= 1) s += __shfl_xor(s, off, 32);
    float scale = 1.f / fmaxf(sqrtf(s), 1e-12f);
    out[wid * 64 + lane]      = v0 * scale;
    out[wid * 64 + 32 + lane] = v1 * scale;
}

// ---------------------------------------------------------------------------
static inline int ceil_div(long long a, long long b) { return (int)((a + b - 1) / b); }

extern "C" void kernel_launch(void* const* d_in, const int* in_sizes, int n_in,
                              void* d_out, int out_size, void* d_ws, size_t ws_size,
                              hipStream_t stream) {
    const float* x    = (const float*)d_in[0];
    const int*   ei1  = (const int*)  d_in[1];
    const int*   ei2  = (const int*)  d_in[2];
    const float* W1   = (const float*)d_in[3];
    const float* as1  = (const float*)d_in[4];
    const float* ad1  = (const float*)d_in[5];
    const float* b1   = (const float*)d_in[6];
    const float* W2   = (const float*)d_in[7];
    const float* as2  = (const float*)d_in[8];
    const float* ad2  = (const float*)d_in[9];
    const float* b2   = (const float*)d_in[10];
    float* out        = (float*)d_out;

    const int EMB = 128, H1 = 4, C1 = 64, D1 = H1 * C1;   // 256
    const int H2 = 1, C2 = 64, D2 = 64;
    const int N  = in_sizes[0] / EMB;                     // 50000
    const int E1 = in_sizes[1] / 2;                       // 800000
    const int E2 = in_sizes[2] / 2;

    // ---- workspace layout (floats) ----
    float* ws     = (float*)d_ws;
    float* h1     = ws;                         // [N, 256]
    float* accum1 = h1 + (size_t)N * D1;        // [N, 256] -> becomes layer-2 input
    float* smalls = accum1 + (size_t)N * D1;
    float*    a_s1   = smalls;                  // [N,4]
    float*    a_d1   = a_s1 + (size_t)N * H1;   // [N,4]
    unsigned* m1     = (unsigned*)(a_d1 + (size_t)N * H1);  // [N,4]
    float*    den1   = (float*)(m1 + (size_t)N * H1);       // [N,4]
    float*    a_s2   = den1 + (size_t)N * H1;   // [N]
    float*    a_d2   = a_s2 + N;                // [N]
    unsigned* m2     = (unsigned*)(a_d2 + N);   // [N]
    float*    den2   = (float*)(m2 + N);        // [N]
    // h1 region reused after layer-1 aggregation:
    float* h2     = h1;                         // [N, 64]
    float* accum2 = h1 + (size_t)N * D2;        // [N, 64] (disjoint from h2)

    const dim3 blk(256);

    // 1) h1 = x @ W1   (M=N, K=128, Ncols=256); 8 waves -> 1x8 tiles of 16x32
    {
        dim3 grid(ceil_div(N, 16), 1);
        size_t lds = (size_t)16 * EMB * sizeof(float);    // 8 KB
        hipLaunchKernelGGL(gat_wmma_gemm_f32, grid, blk, lds, stream,
                           x, W1, h1, N, EMB, D1, 1, 8);
    }
    // 2) attention logits layer 1
    hipLaunchKernelGGL(gat_att_logits, dim3(ceil_div((long long)N * 32, 256)), blk, 0, stream,
                       h1, as1, ad1, a_s1, a_d1, N, H1, C1);
    // 3) init m1/den1/accum1
    hipLaunchKernelGGL(gat_init, dim3(ceil_div((long long)N * D1, 256)), blk, 0, stream,
                       m1, den1, accum1, N * H1, N * D1);
    // 4) segment max
    hipLaunchKernelGGL(gat_edge_max, dim3(ceil_div((long long)(E1 + N) * H1, 256)), blk, 0, stream,
                       ei1, E1, N, H1, a_s1, a_d1, m1);
    // 5) weighted scatter-sum
    hipLaunchKernelGGL(gat_edge_agg, dim3(ceil_div((long long)(E1 + N) * 32, 256)), blk, 0, stream,
                       ei1, E1, N, H1, C1, a_s1, a_d1, m1, h1, den1, accum1);
    // 6) finalize + bias + ReLU (in place -> layer-2 input)
    hipLaunchKernelGGL(gat_finalize, dim3(ceil_div((long long)N * D1, 256)), blk, 0, stream,
                       accum1, den1, b1, N, H1, C1, 1);
    // 7) h2 = relu_out @ W2  (M=N, K=256, Ncols=64); 8 waves -> 4x2 tiles of 16x32
    {
        dim3 grid(ceil_div(N, 64), 1);
        size_t lds = (size_t)64 * D1 * sizeof(float);     // 64 KB (320 KB WGP pool)
        hipLaunchKernelGGL(gat_wmma_gemm_f32, grid, blk, lds, stream,
                           accum1, W2, h2, N, D1, D2, 4, 2);
    }
    // 8) attention logits layer 2
    hipLaunchKernelGGL(gat_att_logits, dim3(ceil_div((long long)N * 32, 256)), blk, 0, stream,
                       h2, as2, ad2, a_s2, a_d2, N, H2, C2);
    // 9) init m2/den2/accum2
    hipLaunchKernelGGL(gat_init, dim3(ceil_div((long long)N * D2, 256)), blk, 0, stream,
                       m2, den2, accum2, N * H2, N * D2);
    // 10) segment max layer 2
    hipLaunchKernelGGL(gat_edge_max, dim3(ceil_div((long long)(E2 + N) * H2, 256)), blk, 0, stream,
                       ei2, E2, N, H2, a_s2, a_d2, m2);
    // 11) weighted scatter-sum layer 2
    hipLaunchKernelGGL(gat_edge_agg, dim3(ceil_div((long long)(E2 + N) * 32, 256)), blk, 0, stream,
                       ei2, E2, N, H2, C2, a_s2, a_d2, m2, h2, den2, accum2);
    // 12) finalize + bias (no relu)
    hipLaunchKernelGGL(gat_finalize, dim3(ceil_div((long long)N * D2, 256)), blk, 0, stream,
                       accum2, den2, b2, N, H2, C2, 0);
    // 13) row-wise L2 normalize -> d_out
    hipLaunchKernelGGL(gat_l2norm, dim3(ceil_div((long long)N * 32, 256)), blk, 0, stream,
                       accum2, out, N);

    (void)n_in; (void)out_size; (void)ws_size;
}